// PointSIFT_module_basic_33071248179391
// MI455X (gfx1250) — compile-verified
//
#include <hip/hip_runtime.h>
#include <stdint.h>

typedef float v2f __attribute__((ext_vector_type(2)));
typedef float v8f __attribute__((ext_vector_type(8)));

#define NPTS     4096
#define BATCH    2
#define FEAT     64
#define RADIUS_F 0.1f
#define TILE     16
#define NTILE    (NPTS / TILE)     /* 256 column tiles */
#define WPB      8                 /* waves per block  */

/* d_out layout (flat float32/int32):
   grouped_xyz  [2,4096,8,3 ]  @ 0
   grouped_pts  [2,4096,8,67]  @ 196608
   idx          [2,4096,8   ]  @ 4587520                         */
#define GP_OFF   (BATCH * NPTS * 8 * 3)
#define IDX_OFF  (GP_OFF + BATCH * NPTS * 8 * (3 + FEAT))

static __device__ __forceinline__ unsigned umin32(unsigned a, unsigned b) {
    return a < b ? a : b;
}

/* ---------------- Kernel 1: octant nearest-neighbor via WMMA Gram matrix -- */
__global__ __launch_bounds__(WPB * 32) void pointsift_select_cube_wmma(
    const float* __restrict__ xyz, int* __restrict__ idx_out)
{
    __shared__ unsigned lds_best[TILE * 8];   /* [row 0..15][oct 0..7] */

    const int tid   = threadIdx.x;
    const int lane  = tid & 31;
    const int wv    = tid >> 5;
    const int b     = blockIdx.x / NTILE;
    const int itile = blockIdx.x % NTILE;
    const int i0    = itile * TILE;
    const float* xb = xyz + (size_t)b * NPTS * 3;

    if (tid < TILE * 8) lds_best[tid] = 0xFFFFFFFFu;
    __syncthreads();

    /* A operand: 16x4 f32 (K = x,y,z,0).
       lanes 0-15 : VGPR0=K0 (x[row]), VGPR1=K1 (y[row])
       lanes16-31 : VGPR0=K2 (z[row]), VGPR1=K3 (0)       row = lane&15 */
    const int arow = i0 + (lane & 15);
    v2f amat;
    {
        const float ax = xb[arow * 3 + 0];
        const float ay = xb[arow * 3 + 1];
        const float az = xb[arow * 3 + 2];
        amat.x = (lane < 16) ? ax : az;
        amat.y = (lane < 16) ? ay : 0.0f;
    }

    /* Query rows this lane owns in the D tile: M = half + k, k=0..7 */
    const int half = (lane >> 4) * 8;
    float qx[8], qy[8], qz[8], ri[8];
    #pragma unroll
    for (int k = 0; k < 8; ++k) {
        const int r = i0 + half + k;
        qx[k] = xb[r * 3 + 0];
        qy[k] = xb[r * 3 + 1];
        qz[k] = xb[r * 3 + 2];
        ri[k] = fmaf(qx[k], qx[k], fmaf(qy[k], qy[k], qz[k] * qz[k]));
    }

    /* Running minima: packed key = (dist_bits & ~0xFFF) | j  (j < 4096).
       Monotone in dist, ties resolved toward smaller j (matches argmin). */
    unsigned best[64];
    #pragma unroll
    for (int s = 0; s < 64; ++s) best[s] = 0xFFFFFFFFu;

    #pragma unroll 2
    for (int jt = wv; jt < NTILE; jt += WPB) {
        const int   jl = jt * TILE + (lane & 15);
        const float xj = xb[jl * 3 + 0];
        const float yj = xb[jl * 3 + 1];
        const float zj = xb[jl * 3 + 2];
        const float rj = fmaf(xj, xj, fmaf(yj, yj, zj * zj));

        /* prefetch next tile's coordinates (global_prefetch_b8) */
        {
            const int jn = (jt + WPB < NTILE) ? jt + WPB : jt;
            __builtin_prefetch(&xb[(jn * TILE + (lane & 15)) * 3], 0, 0);
        }

        /* B operand: 4x16 f32, N = lane&15.
           VGPR0: lanes0-15 K0-row (x), lanes16-31 K1-row (y)
           VGPR1: lanes0-15 K2-row (z), lanes16-31 K3-row (0) */
        v2f bmat;
        bmat.x = (lane < 16) ? xj : yj;
        bmat.y = (lane < 16) ? zj : 0.0f;

        v8f c = {0.f, 0.f, 0.f, 0.f, 0.f, 0.f, 0.f, 0.f};
        v8f dmat = __builtin_amdgcn_wmma_f32_16x16x4_f32(
            false, amat, false, bmat, (short)0, c, false, false);

        #pragma unroll
        for (int k = 0; k < 8; ++k) {
            const float dx = xj - qx[k];
            const float dy = yj - qy[k];
            const float dz = zj - qz[k];
            /* dist = |xi|^2 + |xj|^2 - 2*dot ; clamp cancellation below 0 */
            float dist = fmaxf(fmaf(-2.0f, dmat[k], ri[k] + rj), 0.0f);
            const bool within = (fabsf(dx) < RADIUS_F) &
                                (fabsf(dy) < RADIUS_F) &
                                (fabsf(dz) < RADIUS_F);
            const int oct = ((dx > 0.f) ? 4 : 0) +
                            ((dy > 0.f) ? 2 : 0) +
                            ((dz > 0.f) ? 1 : 0);
            const unsigned key = within
                ? ((__float_as_uint(dist) & 0xFFFFF000u) | (unsigned)jl)
                : 0xFFFFFFFFu;
            #pragma unroll
            for (int o = 0; o < 8; ++o) {
                const unsigned cand = (oct == o) ? key : 0xFFFFFFFFu;
                best[k * 8 + o] = umin32(best[k * 8 + o], cand);
            }
        }
    }

    /* Reduce over the 16 column-lanes of each half (rows stay per-half). */
    #pragma unroll
    for (int s = 0; s < 64; ++s) {
        unsigned v = best[s];
        v = umin32(v, (unsigned)__shfl_xor((int)v, 1, 32));
        v = umin32(v, (unsigned)__shfl_xor((int)v, 2, 32));
        v = umin32(v, (unsigned)__shfl_xor((int)v, 4, 32));
        v = umin32(v, (unsigned)__shfl_xor((int)v, 8, 32));
        best[s] = v;
    }

    /* Cross-wave combine: lane (lane&15)==o (o<8) publishes octant o. */
    const int osel = lane & 15;
    if (osel < 8) {
        #pragma unroll
        for (int k = 0; k < 8; ++k)
            atomicMin(&lds_best[(half + k) * 8 + osel], best[k * 8 + osel]);
    }
    __syncthreads();

    if (tid < TILE * 8) {
        const int m = tid >> 3, o = tid & 7;
        const int i = i0 + m;
        const unsigned key = lds_best[tid];
        const int j = (key == 0xFFFFFFFFu) ? i : (int)(key & 0xFFFu);
        idx_out[((size_t)b * NPTS + i) * 8 + o] = j;
    }
}

/* ---------------- Kernel 2: bandwidth-bound gather ------------------------ */
__global__ __launch_bounds__(64) void pointsift_gather(
    const float* __restrict__ xyz, const float* __restrict__ points,
    const int* __restrict__ idx, float* __restrict__ gx, float* __restrict__ gp)
{
    const int g  = blockIdx.x;      /* (b*N + i)*8 + o */
    const int bi = g >> 3;          /* b*N + i         */
    const int b  = bi / NPTS;
    const int t  = threadIdx.x;     /* 0..63           */
    const int j  = idx[g];
    const size_t jb = (size_t)b * NPTS + j;

    gp[(size_t)g * (3 + FEAT) + 3 + t] = points[jb * FEAT + t];
    if (t < 3) {
        const float d = xyz[jb * 3 + t] - xyz[(size_t)bi * 3 + t];
        gx[(size_t)g * 3 + t]          = d;
        gp[(size_t)g * (3 + FEAT) + t] = d;
    }
}

/* ---------------- Launch -------------------------------------------------- */
extern "C" void kernel_launch(void* const* d_in, const int* in_sizes, int n_in,
                              void* d_out, int out_size, void* d_ws, size_t ws_size,
                              hipStream_t stream)
{
    (void)in_sizes; (void)n_in; (void)out_size; (void)d_ws; (void)ws_size;
    const float* xyz    = (const float*)d_in[0];
    const float* points = (const float*)d_in[1];
    float* out = (float*)d_out;
    int*   idx = (int*)out + IDX_OFF;

    pointsift_select_cube_wmma<<<BATCH * NTILE, WPB * 32, 0, stream>>>(xyz, idx);
    pointsift_gather<<<BATCH * NPTS * 8, 64, 0, stream>>>(
        xyz, points, idx, out, out + GP_OFF);
}